// MAMDense_2834678415865
// MI455X (gfx1250) — compile-verified
//
#include <hip/hip_runtime.h>
#include <cstdint>

#ifndef __has_builtin
#define __has_builtin(x) 0
#endif

typedef float v4f __attribute__((ext_vector_type(4)));

#define BM 64
#define BN 64
#define BK 32
#define BKP 36   // padded LDS row stride (floats): 144B rows -> 16B-aligned, bank-spread

// CDNA5 async global->LDS copy, 16B per lane, tracked by ASYNCcnt.
// vdst operand = LDS byte address in a VGPR (low 32 bits of the flat LDS
// pointer ARE the LDS byte address per the gfx1250 aperture rules).
__device__ __forceinline__ void async_copy_b128(const float* gsrc, const float* ldst) {
  unsigned lds_addr = (unsigned)(uintptr_t)ldst;
  unsigned long long gaddr = (unsigned long long)(uintptr_t)gsrc;
  asm volatile("global_load_async_to_lds_b128 %0, %1, off"
               :: "v"(lds_addr), "v"(gaddr)
               : "memory");
}

__device__ __forceinline__ void wait_async0() {
#if __has_builtin(__builtin_amdgcn_s_wait_asynccnt)
  __builtin_amdgcn_s_wait_asynccnt(0);
#else
  asm volatile("s_wait_asynccnt 0" ::: "memory");
#endif
}

// C[m,n] = max_k(X[m,k]*W[n,k]) + min_k(X[m,k]*W[n,k]) + bias[n]
__global__ __launch_bounds__(256) void mam_dense_kernel(
    const float* __restrict__ X, const float* __restrict__ W,
    const float* __restrict__ bias, float* __restrict__ C,
    int M, int N, int K) {
  __shared__ __align__(16) float As[2][BM * BKP];
  __shared__ __align__(16) float Ws[2][BN * BKP];

  const int tid = threadIdx.x;
  const int tx = tid & 15;   // n direction (16 threads)
  const int ty = tid >> 4;   // m direction (16 threads)
  const int m0 = blockIdx.y * BM;
  const int n0 = blockIdx.x * BN;

  float mx[4][4], mn[4][4];
#pragma unroll
  for (int i = 0; i < 4; ++i)
#pragma unroll
    for (int j = 0; j < 4; ++j) {
      mx[i][j] = -__builtin_inff();
      mn[i][j] =  __builtin_inff();
    }

  const int nTiles = K / BK;

  // Stage one 64x32 tile of X and one 64x32 tile of W (both K-contiguous in
  // memory). 512 16B chunks per tile / 256 threads = 2 chunks each.
  auto issue = [&](int kt, int buf) {
    const int k0 = kt * BK;
#pragma unroll
    for (int c = 0; c < 2; ++c) {
      const int chunk = tid * 2 + c;       // 0..511
      const int row   = chunk >> 3;        // 0..63  (8 chunks per 32-float row)
      const int col   = (chunk & 7) << 2;  // float offset within row
      async_copy_b128(X + (size_t)(m0 + row) * K + k0 + col,
                      &As[buf][row * BKP + col]);
      async_copy_b128(W + (size_t)(n0 + row) * K + k0 + col,
                      &Ws[buf][row * BKP + col]);
    }
  };

  issue(0, 0);
  wait_async0();
  __syncthreads();

  for (int kt = 0; kt < nTiles; ++kt) {
    const int cur = kt & 1;
    if (kt + 1 < nTiles) issue(kt + 1, (kt + 1) & 1);  // overlap copy w/ compute

    const float* as = &As[cur][(ty * 4) * BKP];
    const float* ws = &Ws[cur][(tx * 4) * BKP];

#pragma unroll 2
    for (int kk = 0; kk < BK; kk += 4) {
      v4f a[4], w[4];
#pragma unroll
      for (int i = 0; i < 4; ++i) a[i] = *(const v4f*)(as + i * BKP + kk);
#pragma unroll
      for (int j = 0; j < 4; ++j) w[j] = *(const v4f*)(ws + j * BKP + kk);
#pragma unroll
      for (int i = 0; i < 4; ++i)
#pragma unroll
        for (int j = 0; j < 4; ++j)
#pragma unroll
          for (int e = 0; e < 4; e += 2) {
            // Pairwise reduction: two VOPD-pairable muls, then
            // max(max(.,.),.) / min(min(.,.),.) which LLVM folds into
            // single v_max3_num_f32 / v_min3_num_f32 VOP3 ops.
            float p0 = a[i][e]     * w[j][e];
            float p1 = a[i][e + 1] * w[j][e + 1];
            mx[i][j] = fmaxf(fmaxf(mx[i][j], p0), p1);
            mn[i][j] = fminf(fminf(mn[i][j], p0), p1);
          }
    }

    wait_async0();     // this wave's async LDS writes complete
    __syncthreads();   // publish across the workgroup; safe to reuse buffers
  }

  const int nc = n0 + tx * 4;
  v4f b = *(const v4f*)(bias + nc);
#pragma unroll
  for (int i = 0; i < 4; ++i) {
    const int m = m0 + ty * 4 + i;
    v4f o;
#pragma unroll
    for (int j = 0; j < 4; ++j) o[j] = mx[i][j] + mn[i][j] + b[j];
    *(v4f*)(C + (size_t)m * N + nc) = o;
  }
}

extern "C" void kernel_launch(void* const* d_in, const int* in_sizes, int n_in,
                              void* d_out, int out_size, void* d_ws, size_t ws_size,
                              hipStream_t stream) {
  (void)n_in; (void)out_size; (void)d_ws; (void)ws_size;
  const float* X = (const float*)d_in[0];   // [M,K] f32
  const float* W = (const float*)d_in[1];   // [N,K] f32
  const float* b = (const float*)d_in[2];   // [N]   f32
  float* C = (float*)d_out;                 // [M,N] f32

  const int N = in_sizes[2];
  const int K = in_sizes[1] / N;
  const int M = in_sizes[0] / K;

  dim3 grid(N / BN, M / BM);
  mam_dense_kernel<<<grid, dim3(256), 0, stream>>>(X, W, b, C, M, N, K);
}